// GloVeLoss_1056561955284
// MI455X (gfx1250) — compile-verified
//
#include <hip/hip_runtime.h>
#include <math.h>

typedef __attribute__((ext_vector_type(2))) float v2f;
typedef __attribute__((ext_vector_type(8))) float v8f;

#define BLOCK 256
#define NBLK  512

// Main gather + square-error accumulate. Memory-latency bound (random 4B
// gathers over a 256MB table); 4096 waves in flight to hide latency.
__global__ void glove_partial(const float* __restrict__ counts,
                              const int*   __restrict__ gram,
                              const int*   __restrict__ context,
                              const float* __restrict__ pred,
                              float*       __restrict__ partials,
                              int NK, int V, int K) {
  int tid    = blockIdx.x * blockDim.x + threadIdx.x;
  int stride = gridDim.x * blockDim.x;

  float acc = 0.0f;
  for (int idx = tid; idx < NK; idx += stride) {
    int g = gram[idx / K];              // 8 consecutive lanes share one entry
    int c = context[idx];               // coalesced
    float cnt = counts[(size_t)g * (size_t)V + (size_t)c];  // random gather
    float p   = pred[idx];              // coalesced
    float gt  = log1pf(cnt);
    float d   = p - gt;
    acc += (cnt > 0.0f) ? d * d : 0.0f; // presence-mask weighting (bug-faithful)
  }

  // --- wave32 cross-lane reduction via V_WMMA_F32_16X16X4_F32 (full fp32) ---
  // A is 16x4 f32: lane L<16 holds A[L][0..1] in 2 VGPRs, lane L>=16 holds
  // A[L-16][2..3]. With a = {acc, 0} per lane, row m of A sums to
  // acc[m] + acc[m+16]. B = ones(4x16)  =>  D[m][n] = rowsum(m).
  // Each lane's 8 D regs are 8 distinct row-sums: lanes 0-15 see rows 0-7,
  // lanes 16-31 see rows 8-15. Sum them, then one xor-16 shuffle gives the
  // exact 32-lane fp32 sum in every lane.
  v2f a; a.x = acc;  a.y = 0.0f;
  v2f b; b.x = 1.0f; b.y = 1.0f;
  v8f cm = {0.0f, 0.0f, 0.0f, 0.0f, 0.0f, 0.0f, 0.0f, 0.0f};
  v8f dm = __builtin_amdgcn_wmma_f32_16x16x4_f32(false, a, false, b,
                                                 (short)0, cm, false, false);
  float s = dm[0] + dm[1] + dm[2] + dm[3] + dm[4] + dm[5] + dm[6] + dm[7];
  s += __shfl_xor(s, 16, 32);

  __shared__ float wsum[BLOCK / 32];
  int lane = threadIdx.x & 31;
  int wave = threadIdx.x >> 5;
  if (lane == 0) wsum[wave] = s;
  __syncthreads();
  if (threadIdx.x == 0) {
    float t = 0.0f;
#pragma unroll
    for (int i = 0; i < BLOCK / 32; ++i) t += wsum[i];
    partials[blockIdx.x] = t;           // deterministic slot write, no atomics
  }
}

// Reduce the NBLK block partials and write the mean.
__global__ void glove_finalize(const float* __restrict__ partials,
                               float* __restrict__ out, int nb, int NK) {
  __shared__ float sh[BLOCK];
  float s = 0.0f;
  for (int i = threadIdx.x; i < nb; i += BLOCK) s += partials[i];
  sh[threadIdx.x] = s;
  __syncthreads();
  for (int off = BLOCK / 2; off > 0; off >>= 1) {
    if (threadIdx.x < off) sh[threadIdx.x] += sh[threadIdx.x + off];
    __syncthreads();
  }
  if (threadIdx.x == 0) out[0] = sh[0] / (float)NK;
}

extern "C" void kernel_launch(void* const* d_in, const int* in_sizes, int n_in,
                              void* d_out, int out_size, void* d_ws, size_t ws_size,
                              hipStream_t stream) {
  const float* counts  = (const float*)d_in[0];   // (V, V) fp32
  const int*   gram    = (const int*)  d_in[1];   // (N, 1) int32
  const int*   context = (const int*)  d_in[2];   // (N, K) int32
  const float* pred    = (const float*)d_in[3];   // (N, K) fp32

  float* out      = (float*)d_out;                // scalar
  float* partials = (float*)d_ws;                 // NBLK floats of scratch

  int NK = in_sizes[3];                           // N*K
  int N  = in_sizes[1];
  int K  = (N > 0) ? (NK / N) : 1;
  int V  = (int)(sqrt((double)in_sizes[0]) + 0.5);

  glove_partial<<<NBLK, BLOCK, 0, stream>>>(counts, gram, context, pred,
                                            partials, NK, V, K);
  glove_finalize<<<1, BLOCK, 0, stream>>>(partials, out, NBLK, NK);
}